// Message_70738111365649
// MI455X (gfx1250) — compile-verified
//
#include <hip/hip_runtime.h>
#include <hip/hip_bf16.h>
#include <math.h>

// ---------------- problem constants (match reference) ----------------
#define SDIM 128                // memory / hidden dim
#define TDIM 64                 // time-encoding dim
#define EDIM 64                 // edge feature dim
#define MDIM 384                // 2*S + T + ED

typedef __attribute__((ext_vector_type(2))) float v2f;
typedef __attribute__((ext_vector_type(8))) float v8f;

// -0.32894072 = -9*ln(10)/63 ; freq[i] = 10^(-9*i/63) = exp(-0.32894072*i)
#define FREQ_LN (-0.32894072234f)

// ---------------------------------------------------------------------
// Message phase: one block (128 threads) per edge.
// mail = [src_mem(128) | dst_mem(128) | cos(ts*freq)(64) | e(64)]
// atomically accumulated into sum[dst, 0:384]; count per dst.
// Atomics land in L2 (sum buffers = 2 x 76.8 MB < 192 MB L2).
// ---------------------------------------------------------------------
__global__ __launch_bounds__(128)
void edge_msg_kernel(const float* __restrict__ smem,   // src-side memory  [*,128]
                     const float* __restrict__ dmem,   // dst-side memory  [*,128]
                     const float* __restrict__ ts,     // [E]
                     const float* __restrict__ ef,     // [E,64]
                     const int*   __restrict__ src,    // [E]
                     const int*   __restrict__ dst,    // [E]
                     float* __restrict__ sum,          // [Ndst,384]
                     float* __restrict__ cnt)          // [Ndst]
{
    const int e   = blockIdx.x;
    const int tid = threadIdx.x;            // 0..127
    const int s   = src[e];
    const int d   = dst[e];
    float* o = sum + (size_t)d * MDIM;

    atomicAdd(o + tid,        smem[(size_t)s * SDIM + tid]);       // mail[0:128]
    atomicAdd(o + SDIM + tid, dmem[(size_t)d * SDIM + tid]);       // mail[128:256]
    if (tid < TDIM) {                                              // mail[256:320]
        float f = __expf(FREQ_LN * (float)tid);
        atomicAdd(o + 2 * SDIM + tid, __cosf(ts[e] * f));
    } else {                                                       // mail[320:384]
        atomicAdd(o + 2 * SDIM + tid, ef[(size_t)e * EDIM + (tid - TDIM)]);
    }
    if (tid == 0) atomicAdd(cnt + d, 1.0f);
}

// ---------------------------------------------------------------------
// Fused mean + GRU step.
// Block: 256 threads = 8 waves; 32 rows x 128 output cols per block.
// Wave w -> output cols [16w,16w+16), two 16-row tiles reusing the same
// B fragments (halves L2 weight traffic, 6 WMMA per 3 global loads).
// B loads software-pipelined one K-step ahead to avoid loadcnt==0 stalls.
// ---------------------------------------------------------------------
__global__ __launch_bounds__(256)
void gru_kernel(const float* __restrict__ sum,    // [N,384] segment sums
                const float* __restrict__ cnt,    // [N]
                const float* __restrict__ h_in,   // [N,128]
                const float* __restrict__ w_ih,   // [384,384]
                const float* __restrict__ w_hh,   // [384,128]
                const float* __restrict__ b_ih,   // [384]
                const float* __restrict__ b_hh,   // [384]
                float* __restrict__ out,          // [N,128]
                int nrows)
{
    __shared__ float xs[32][MDIM + 4];   // pad 4: rows hit distinct banks
    __shared__ float hs[32][SDIM + 4];
    __shared__ float scl[32];

    const int row0 = blockIdx.x * 32;
    const int tid  = threadIdx.x;
    const int lane = tid & 31;
    const int wave = tid >> 5;            // 0..7
    const int c    = wave * 16;           // output col base

    if (tid < 32) {
        int r = row0 + tid; if (r >= nrows) r = nrows - 1;
        scl[tid] = 1.0f / fmaxf(cnt[r], 1.0f);
    }
    __syncthreads();

    // stage mean messages (scaled) and hidden state in LDS, coalesced;
    // clamp row index for the tail block (keeps EXEC全 for WMMA later)
    for (int i = tid; i < 32 * MDIM; i += 256) {
        int r = i / MDIM, k = i - r * MDIM;
        int rr = row0 + r; if (rr >= nrows) rr = nrows - 1;
        xs[r][k] = sum[(size_t)rr * MDIM + k] * scl[r];
    }
    for (int i = tid; i < 32 * SDIM; i += 256) {
        int r = i >> 7, k = i & (SDIM - 1);
        int rr = row0 + r; if (rr >= nrows) rr = nrows - 1;
        hs[r][k] = h_in[(size_t)rr * SDIM + k];
    }
    __syncthreads();

    const int n    = lane & 15;           // N index within tile
    const int koff = (lane >> 4) << 1;    // 0 for lanes 0-15, 2 for 16-31
    const int m16  = lane & 15;           // A-fragment row (within tile)

    // seed accumulators with biases (bias depends only on N = lane&15)
    const float br = b_ih[c + n]            + b_hh[c + n];
    const float bz = b_ih[SDIM + c + n]     + b_hh[SDIM + c + n];
    const float bi = b_ih[2 * SDIM + c + n];
    const float bh = b_hh[2 * SDIM + c + n];
    v8f ar[2], az[2], ai[2], ah[2];
    #pragma unroll
    for (int t = 0; t < 2; ++t)
        #pragma unroll
        for (int i = 0; i < 8; ++i) {
            ar[t][i] = br; az[t][i] = bz; ai[t][i] = bi; ah[t][i] = bh;
        }

    // per-lane weight row pointers (B[k,n] = W[o = base+n, k])
    const float* wr = w_ih + (size_t)(c + n) * MDIM;
    const float* wz = w_ih + (size_t)(SDIM + c + n) * MDIM;
    const float* wn = w_ih + (size_t)(2 * SDIM + c + n) * MDIM;

    // ---- gi = x @ W_ih^T : K = 384, pipelined B loads ----
    v2f b0, b1, b2;
    b0[0] = wr[koff]; b0[1] = wr[koff + 1];
    b1[0] = wz[koff]; b1[1] = wz[koff + 1];
    b2[0] = wn[koff]; b2[1] = wn[koff + 1];
    #pragma unroll 4
    for (int k0 = 0; k0 < MDIM; k0 += 4) {
        const int kn = (k0 + 4 < MDIM) ? (k0 + 4 + koff) : koff;
        v2f nb0; nb0[0] = wr[kn]; nb0[1] = wr[kn + 1];
        v2f nb1; nb1[0] = wz[kn]; nb1[1] = wz[kn + 1];
        v2f nb2; nb2[0] = wn[kn]; nb2[1] = wn[kn + 1];
        v2f a0;  a0[0]  = xs[m16][k0 + koff];      a0[1] = xs[m16][k0 + koff + 1];
        v2f a1;  a1[0]  = xs[m16 + 16][k0 + koff]; a1[1] = xs[m16 + 16][k0 + koff + 1];
        ar[0] = __builtin_amdgcn_wmma_f32_16x16x4_f32(false, a0, false, b0, (short)0, ar[0], false, false);
        ar[1] = __builtin_amdgcn_wmma_f32_16x16x4_f32(false, a1, false, b0, (short)0, ar[1], false, false);
        az[0] = __builtin_amdgcn_wmma_f32_16x16x4_f32(false, a0, false, b1, (short)0, az[0], false, false);
        az[1] = __builtin_amdgcn_wmma_f32_16x16x4_f32(false, a1, false, b1, (short)0, az[1], false, false);
        ai[0] = __builtin_amdgcn_wmma_f32_16x16x4_f32(false, a0, false, b2, (short)0, ai[0], false, false);
        ai[1] = __builtin_amdgcn_wmma_f32_16x16x4_f32(false, a1, false, b2, (short)0, ai[1], false, false);
        b0 = nb0; b1 = nb1; b2 = nb2;
    }

    // ---- gh = h @ W_hh^T : K = 128, pipelined B loads ----
    const float* vr = w_hh + (size_t)(c + n) * SDIM;
    const float* vz = w_hh + (size_t)(SDIM + c + n) * SDIM;
    const float* vn = w_hh + (size_t)(2 * SDIM + c + n) * SDIM;
    b0[0] = vr[koff]; b0[1] = vr[koff + 1];
    b1[0] = vz[koff]; b1[1] = vz[koff + 1];
    b2[0] = vn[koff]; b2[1] = vn[koff + 1];
    #pragma unroll 4
    for (int k0 = 0; k0 < SDIM; k0 += 4) {
        const int kn = (k0 + 4 < SDIM) ? (k0 + 4 + koff) : koff;
        v2f nb0; nb0[0] = vr[kn]; nb0[1] = vr[kn + 1];
        v2f nb1; nb1[0] = vz[kn]; nb1[1] = vz[kn + 1];
        v2f nb2; nb2[0] = vn[kn]; nb2[1] = vn[kn + 1];
        v2f a0;  a0[0]  = hs[m16][k0 + koff];      a0[1] = hs[m16][k0 + koff + 1];
        v2f a1;  a1[0]  = hs[m16 + 16][k0 + koff]; a1[1] = hs[m16 + 16][k0 + koff + 1];
        ar[0] = __builtin_amdgcn_wmma_f32_16x16x4_f32(false, a0, false, b0, (short)0, ar[0], false, false);
        ar[1] = __builtin_amdgcn_wmma_f32_16x16x4_f32(false, a1, false, b0, (short)0, ar[1], false, false);
        az[0] = __builtin_amdgcn_wmma_f32_16x16x4_f32(false, a0, false, b1, (short)0, az[0], false, false);
        az[1] = __builtin_amdgcn_wmma_f32_16x16x4_f32(false, a1, false, b1, (short)0, az[1], false, false);
        ah[0] = __builtin_amdgcn_wmma_f32_16x16x4_f32(false, a0, false, b2, (short)0, ah[0], false, false);
        ah[1] = __builtin_amdgcn_wmma_f32_16x16x4_f32(false, a1, false, b2, (short)0, ah[1], false, false);
        b0 = nb0; b1 = nb1; b2 = nb2;
    }

    // fused GRU gates (D layout: VGPR i -> M = i + 8*(lane>=16), N = lane&15)
    const int mbase = (lane >> 4) << 3;
    #pragma unroll
    for (int t = 0; t < 2; ++t) {
        #pragma unroll
        for (int i = 0; i < 8; ++i) {
            int   m  = 16 * t + mbase + i;
            float r  = 1.0f / (1.0f + __expf(-ar[t][i]));
            float z  = 1.0f / (1.0f + __expf(-az[t][i]));
            float nn = tanhf(ai[t][i] + r * ah[t][i]);
            float h  = hs[m][c + n];
            int row  = row0 + m;
            if (row < nrows)
                out[(size_t)row * SDIM + c + n] = (1.0f - z) * nn + z * h;
        }
    }
}

// ---------------------------------------------------------------------
extern "C" void kernel_launch(void* const* d_in, const int* in_sizes, int n_in,
                              void* d_out, int out_size, void* d_ws, size_t ws_size,
                              hipStream_t stream) {
    const float* si   = (const float*)d_in[0];
    const float* sj   = (const float*)d_in[1];
    const float* si_r = (const float*)d_in[2];
    const float* sj_r = (const float*)d_in[3];
    const float* t    = (const float*)d_in[4];
    const float* t_r  = (const float*)d_in[5];
    const float* e    = (const float*)d_in[6];
    const float* e_r  = (const float*)d_in[7];
    const float* w_ih = (const float*)d_in[8];
    const float* w_hh = (const float*)d_in[9];
    const float* b_ih = (const float*)d_in[10];
    const float* b_hh = (const float*)d_in[11];
    const int* src_g  = (const int*)d_in[12];
    const int* dst_g  = (const int*)d_in[13];
    const int* src_gr = (const int*)d_in[14];
    const int* dst_gr = (const int*)d_in[15];
    float* out = (float*)d_out;

    const int E       = in_sizes[4];            // 400000
    const int n_users = in_sizes[0] / SDIM;     // 50000
    const int n_items = in_sizes[1] / SDIM;     // 50000

    // workspace layout (fp32): sum_j | sum_i | cnt_j | cnt_i
    float* sum_j = (float*)d_ws;
    float* sum_i = sum_j + (size_t)n_items * MDIM;
    float* cnt_j = sum_i + (size_t)n_users * MDIM;
    float* cnt_i = cnt_j + n_items;
    size_t zero_bytes = ((size_t)(n_items + n_users) * MDIM
                         + (size_t)(n_items + n_users)) * sizeof(float);
    hipMemsetAsync(d_ws, 0, zero_bytes, stream);

    // messages to items (graph g) and to users (graph g_r)
    edge_msg_kernel<<<E, 128, 0, stream>>>(si,   sj,   t,   e,   src_g,  dst_g,  sum_j, cnt_j);
    edge_msg_kernel<<<E, 128, 0, stream>>>(si_r, sj_r, t_r, e_r, src_gr, dst_gr, sum_i, cnt_i);

    // si_out = GRU(mean_i, sj_r) ; sj_out = GRU(mean_j, sj)
    const int gb_u = (n_users + 31) / 32;
    const int gb_i = (n_items + 31) / 32;
    gru_kernel<<<gb_u, 256, 0, stream>>>(sum_i, cnt_i, sj_r,
                                         w_ih, w_hh, b_ih, b_hh, out, n_users);
    gru_kernel<<<gb_i, 256, 0, stream>>>(sum_j, cnt_j, sj,
                                         w_ih, w_hh, b_ih, b_hh,
                                         out + (size_t)n_users * SDIM, n_items);
}